// SudokuIterate_12446815224332
// MI455X (gfx1250) — compile-verified
//
#include <hip/hip_runtime.h>
#include <hip/hip_bf16.h>
#include <stdint.h>

// ---------------------------------------------------------------------------
// Sudoku iterate: B boards of [9,9,9] f32. Memory-bound copy-with-one-patch.
// CDNA5 path: async global->LDS staging (ASYNCcnt) so sudoku is read from HBM
// exactly once while serving both the selection reduction and the copy-out.
// All streams are touched exactly once -> non-temporal hints end to end.
// 1 wave32 per board, 8 boards / 256-thread block.
// ---------------------------------------------------------------------------

#define KSZ 9
#define NCELLS 81
#define NELEM 729                 // 9*9*9 floats per board
#define BOARDS_PER_BLOCK 8
#define LDS_STRIDE 768            // padded slot (floats), 3072 B per board

__device__ __forceinline__ void async_b32_to_lds_nt(uint32_t lds_byte_off,
                                                    const float* gaddr) {
    uint64_t a = (uint64_t)(uintptr_t)gaddr;
    // CDNA5 VGLOBAL async op: copies one dword from global memory into LDS at VDST; tracked by ASYNCcnt.
    // NT temporal hint: stream, read-once, don't pollute L2.
    asm volatile("global_load_async_to_lds_b32 %0, %1, off th:TH_LOAD_NT"
                 :: "v"(lds_byte_off), "v"(a)
                 : "memory");
}

__global__ __launch_bounds__(256) void sudoku_iterate_kernel(
    const float* __restrict__ sudoku,     // [B,9,9,9]
    const float* __restrict__ rmask,      // [B,9,9,9]
    const float* __restrict__ rindex,     // [B]
    float* __restrict__ out,              // [B*729 | B*729 | B]
    int B)
{
    __shared__ float lds[BOARDS_PER_BLOCK * LDS_STRIDE];

    const int wave = threadIdx.x >> 5;
    const int lane = threadIdx.x & 31;
    const int b = blockIdx.x * BOARDS_PER_BLOCK + wave;
    if (b >= B) return;

    float* slot = &lds[wave * LDS_STRIDE];
    // Low 32 bits of a generic pointer into LDS == workgroup-relative LDS byte offset.
    const uint32_t slot_off = (uint32_t)(uintptr_t)slot;
    const float* gboard = sudoku + (size_t)b * NELEM;

    // ---- Phase 0: async-copy this wave's board into its private LDS slot ----
    #pragma unroll
    for (int k = 0; k < 23; ++k) {
        int e = lane + 32 * k;
        if (e < NELEM) {
            async_b32_to_lds_nt(slot_off + 4u * (uint32_t)e, gboard + e);
        }
    }
    asm volatile("s_wait_asynccnt 0x0" ::: "memory");

    // ---- Phase A: per-cell score + channel argmax, then wave reduction ------
    float best_s = -1.0e30f;
    int   best_q = 0;
    int   best_c = 0;
    float best_v = 0.0f;

    #pragma unroll
    for (int k = 0; k < 3; ++k) {
        int q = lane + 32 * k;               // cell index h*9+w, row-major
        if (q < NCELLS) {
            float sum  = 0.0f;
            float vmax = -1.0e30f;
            int   cmax = 0;
            #pragma unroll
            for (int c = 0; c < KSZ; ++c) {
                float v = slot[c * NCELLS + q];
                sum += v;
                if (v > vmax) { vmax = v; cmax = c; }   // first-max (strict >)
            }
            // Exact reference arithmetic:
            // n = relu(sum-1); n = relu(1-|n|)*(-9) - n
            float n1    = fmaxf(sum - 1.0f, 0.0f);
            float cmp   = fmaxf(1.0f - fabsf(n1), 0.0f);
            float score = cmp * (-(float)KSZ) - n1;
            if (score > best_s) {            // q ascending per lane -> first-max
                best_s = score; best_q = q; best_c = cmax; best_v = vmax;
            }
        }
    }

    // Butterfly reduction: max score, tie -> lowest q (jnp.argmax semantics).
    #pragma unroll
    for (int m = 16; m >= 1; m >>= 1) {
        float s2 = __shfl_xor(best_s, m, 32);
        int   q2 = __shfl_xor(best_q, m, 32);
        int   c2 = __shfl_xor(best_c, m, 32);
        float v2 = __shfl_xor(best_v, m, 32);
        if (s2 > best_s || (s2 == best_s && q2 < best_q)) {
            best_s = s2; best_q = q2; best_c = c2; best_v = v2;
        }
    }

    // ---- Phase B: streamed copy with single-element patch -------------------
    const int   pq        = best_c * NCELLS + best_q;       // flat [c,h,w] index
    const float ridx      = rindex[b];
    const float patch_s   = best_v * (1.0f - best_v);       // sudoku[p]*(1-maxv)
    const float patch_add = ridx * best_v;

    float*       out_sud = out + (size_t)b * NELEM;
    float*       out_rm  = out + (size_t)B * NELEM + (size_t)b * NELEM;
    const float* grm     = rmask + (size_t)b * NELEM;

    #pragma unroll
    for (int k = 0; k < 23; ++k) {
        int e = lane + 32 * k;
        if (e < NELEM) {
            float v  = slot[e];
            __builtin_nontemporal_store((e == pq) ? patch_s : v, out_sud + e);
            float rm = __builtin_nontemporal_load(grm + e);
            __builtin_nontemporal_store((e == pq) ? (rm + patch_add) : rm,
                                        out_rm + e);
        }
    }

    if (lane == 0) {
        out[(size_t)2 * (size_t)B * NELEM + (size_t)b] = ridx + 1.0f;
    }
}

extern "C" void kernel_launch(void* const* d_in, const int* in_sizes, int n_in,
                              void* d_out, int out_size, void* d_ws, size_t ws_size,
                              hipStream_t stream) {
    (void)n_in; (void)out_size; (void)d_ws; (void)ws_size;
    const float* sudoku = (const float*)d_in[0];
    const float* rmask  = (const float*)d_in[1];
    const float* rindex = (const float*)d_in[2];
    float* out = (float*)d_out;

    const int B = in_sizes[0] / NELEM;
    const int blocks = (B + BOARDS_PER_BLOCK - 1) / BOARDS_PER_BLOCK;
    sudoku_iterate_kernel<<<blocks, 256, 0, stream>>>(sudoku, rmask, rindex, out, B);
}